// NIMO_63264868270309
// MI455X (gfx1250) — compile-verified
//
#include <hip/hip_runtime.h>

typedef __attribute__((ext_vector_type(2))) float v2f;
typedef __attribute__((ext_vector_type(8))) float v8f;

#define NS   4096   // samples
#define DIM  128    // input dim (= pe dim)
#define HID  256    // hidden dim
#define OUTC 129    // output columns (1 + DIM)

// ---------------------------------------------------------------------------
// Kernel 0: cpe[j][h] = W1[128+j][h] + b1[h]   (128 x 256)
// ---------------------------------------------------------------------------
__global__ __launch_bounds__(256) void nimo_cpe_kernel(
    const float* __restrict__ W1, const float* __restrict__ b1,
    float* __restrict__ cpe) {
  int idx = blockIdx.x * 256 + threadIdx.x;   // 0 .. 128*256-1
  cpe[idx] = W1[DIM * HID + idx] + b1[idx & (HID - 1)];
}

// ---------------------------------------------------------------------------
// Kernel 1: base = x (4096x128) @ W1[:128] (128x256), fp32 WMMA 16x16x4.
// One wave -> 16(M) x 64(N) tile; A fragment reused across 4 WMMAs per K-step.
// A 16x4 layout (ISA 7.12.2): lanes 0-15 get K=0,1; lanes 16-31 get K=2,3.
// B 4x16 layout mirrors A; C/D: VGPR r = rows {r, r+8}, N = lane&15.
// ---------------------------------------------------------------------------
__global__ __launch_bounds__(256) void nimo_base_gemm_kernel(
    const float* __restrict__ x, const float* __restrict__ W1,
    float* __restrict__ base) {
  const int wave = (blockIdx.x * 256 + threadIdx.x) >> 5;  // 0..1023
  const int lane = threadIdx.x & 31;
  const int mt = wave >> 2;          // 0..255  (M tiles of 16)
  const int nt = wave & 3;           // 0..3    (N tiles of 64)
  const int m0 = mt * 16;
  const int n0 = nt * 64;
  const int lo = lane & 15;
  const int hi = lane >> 4;

  v8f acc0 = {}, acc1 = {}, acc2 = {}, acc3 = {};
  const float* xrow = x + (m0 + lo) * DIM + 2 * hi;

  for (int k = 0; k < DIM; k += 4) {
    v2f a;
    a.x = xrow[k];
    a.y = xrow[k + 1];
    const float* w = W1 + (k + 2 * hi) * HID + n0 + lo;
    v2f bb0, bb1, bb2, bb3;
    bb0.x = w[0];   bb0.y = w[HID];
    bb1.x = w[16];  bb1.y = w[16 + HID];
    bb2.x = w[32];  bb2.y = w[32 + HID];
    bb3.x = w[48];  bb3.y = w[48 + HID];
    acc0 = __builtin_amdgcn_wmma_f32_16x16x4_f32(false, a, false, bb0,
                                                 (short)0, acc0, false, false);
    acc1 = __builtin_amdgcn_wmma_f32_16x16x4_f32(false, a, false, bb1,
                                                 (short)0, acc1, false, false);
    acc2 = __builtin_amdgcn_wmma_f32_16x16x4_f32(false, a, false, bb2,
                                                 (short)0, acc2, false, false);
    acc3 = __builtin_amdgcn_wmma_f32_16x16x4_f32(false, a, false, bb3,
                                                 (short)0, acc3, false, false);
  }

#pragma unroll
  for (int r = 0; r < 8; ++r) {
    int row = m0 + r + 8 * hi;
    float* o = base + row * HID + n0 + lo;
    o[0]  = acc0[r];
    o[16] = acc1[r];
    o[32] = acc2[r];
    o[48] = acc3[r];
  }
}

// ---------------------------------------------------------------------------
// Kernel 2: fused rank-1 update + ReLU + dot(W2) + output assembly.
// Workgroup = 1024 threads (32 waves) handles 32 samples x all 128 features.
// base[n0:n0+32, :] staged in LDS (row stride 257 -> conflict-free wave32
// reads: addr = lane*257 + h, 257 % 64 == 1). Each wave: n = n0+lane, 4 j's.
//   g(n,j) = sum_h relu(base[n,h] - x[n,j]*W1[j,h] + cpe[j,h]) * W2[h] + b2
//   out[n, 1+j] = x[n,j] * (1 + g);  out[n, 0] = 1
// ---------------------------------------------------------------------------
__global__ __launch_bounds__(1024) void nimo_gate_kernel(
    const float* __restrict__ x, const float* __restrict__ base,
    const float* __restrict__ W1, const float* __restrict__ cpe,
    const float* __restrict__ W2, const float* __restrict__ b2,
    float* __restrict__ out) {
  __shared__ float lds_base[32 * 257];
  __shared__ float lds_w2[HID];

  const int tid = threadIdx.x;
  const int n0 = blockIdx.x * 32;

  // stage base tile (32 x 256), padded rows
  for (int i = tid; i < 32 * HID; i += 1024) {
    int n = i >> 8;        // /HID
    int h = i & (HID - 1);
    lds_base[n * 257 + h] = base[(n0 + n) * HID + h];
  }
  if (tid < HID) lds_w2[tid] = W2[tid];
  __syncthreads();

  const int lane = tid & 31;
  const int wave = tid >> 5;         // 0..31
  const int n = n0 + lane;
  const float* bp = lds_base + lane * 257;
  const float bias2 = b2[0];

  if (wave == 0) out[n * OUTC] = 1.0f;   // leading ones column

#pragma unroll
  for (int q = 0; q < 4; ++q) {
    const int j = wave * 4 + q;                 // 0..127, wave-uniform
    const float xnj = x[n * DIM + j];
    const float* __restrict__ w1j = W1 + j * HID;   // wave-uniform rows
    const float* __restrict__ cj  = cpe + j * HID;
    float acc = 0.0f;
#pragma unroll 8
    for (int h = 0; h < HID; ++h) {
      float t = fmaf(-xnj, w1j[h], bp[h] + cj[h]);
      acc = fmaf(fmaxf(t, 0.0f), lds_w2[h], acc);
    }
    out[n * OUTC + 1 + j] = xnj * (1.0f + acc + bias2);
  }
}

// ---------------------------------------------------------------------------
extern "C" void kernel_launch(void* const* d_in, const int* in_sizes, int n_in,
                              void* d_out, int out_size, void* d_ws, size_t ws_size,
                              hipStream_t stream) {
  const float* x  = (const float*)d_in[0];   // (4096,128)
  const float* W1 = (const float*)d_in[1];   // (256,256)
  const float* b1 = (const float*)d_in[2];   // (256,)
  const float* W2 = (const float*)d_in[3];   // (256,1)
  const float* b2 = (const float*)d_in[4];   // (1,)
  float* out = (float*)d_out;                // (4096,129)

  float* base = (float*)d_ws;                // 4096*256 floats (4 MB)
  float* cpe  = base + NS * HID;             // 128*256 floats (128 KB)

  nimo_cpe_kernel<<<dim3((DIM * HID) / 256), dim3(256), 0, stream>>>(W1, b1, cpe);
  nimo_base_gemm_kernel<<<dim3(128), dim3(256), 0, stream>>>(x, W1, base);
  nimo_gate_kernel<<<dim3(NS / 32), dim3(1024), 0, stream>>>(x, base, W1, cpe,
                                                             W2, b2, out);
}